// model_arrival_34084860461716
// MI455X (gfx1250) — compile-verified
//
#include <hip/hip_runtime.h>
#include <hip/hip_bf16.h>

// ---------------------------------------------------------------------------
// EV-charging simulation (E=1024 cohorts, T=1024 steps, S samples).
// Latency-bound sequential scan: keep all state in VGPRs, one wave32 per
// Monte-Carlo sample, and do the per-step cross-lane reduction with exact-f32
// V_WMMA_F32_16X16X4_F32 chains against a ones matrix (CDNA5 matrix pipe,
// co-executes with VALU; no LDS/barriers in the hot loop).
// ---------------------------------------------------------------------------

typedef __attribute__((ext_vector_type(2))) float v2f;
typedef __attribute__((ext_vector_type(8))) float v8f;

#define E_DIM 1024
#define T_DIM 1024
#define N_TOT 100000.0f
#define P_CAP 8000.0f
#define U0_RATE 0.6f
#define FINAL_E 7.2f   // 0.9 * battery_capacity(8.0)

// Exact f32 sum of one value per lane across a wave32, via WMMA.
// Stage 1: A = [p, 0] (16x4), B = ones(4x16)  ->  D1 vgpr d holds
//          r[d] (lanes 0-15) / r[8+d] (lanes 16-31), r[m] = p[m]+p[m+16].
// Stage 2: reinterpret D1 vgpr pairs as A operands; four accumulating WMMAs
//          add r[0..15] -> every lane of acc holds the full 32-lane sum.
__device__ __forceinline__ float wave_sum_wmma(float v) {
  v2f a = {v, 0.0f};
  v2f ones = {1.0f, 1.0f};
  v8f z = {};
  v8f r1 = __builtin_amdgcn_wmma_f32_16x16x4_f32(
      false, a, false, ones, (short)0, z, false, false);
  v8f acc = {};
  v2f a0 = {r1[0], r1[1]};
  acc = __builtin_amdgcn_wmma_f32_16x16x4_f32(
      false, a0, false, ones, (short)0, acc, false, false);
  v2f a1 = {r1[2], r1[3]};
  acc = __builtin_amdgcn_wmma_f32_16x16x4_f32(
      false, a1, false, ones, (short)0, acc, false, false);
  v2f a2 = {r1[4], r1[5]};
  acc = __builtin_amdgcn_wmma_f32_16x16x4_f32(
      false, a2, false, ones, (short)0, acc, false, false);
  v2f a3 = {r1[6], r1[7]};
  acc = __builtin_amdgcn_wmma_f32_16x16x4_f32(
      false, a3, false, ones, (short)0, acc, false, false);
  return acc[0];
}

// n[j] = softmax(y)[j] * N_TOT, one 1024-thread block, LDS tree reductions.
__global__ __launch_bounds__(1024) void softmax_n_kernel(
    const float* __restrict__ y, float* __restrict__ n_out) {
  __shared__ float red[E_DIM];
  const int t = threadIdx.x;
  const float v = y[t];

  red[t] = v;
  __syncthreads();
  for (int off = E_DIM / 2; off > 0; off >>= 1) {
    if (t < off) red[t] = fmaxf(red[t], red[t + off]);
    __syncthreads();
  }
  const float m = red[0];
  __syncthreads();

  const float e = expf(v - m);
  red[t] = e;
  __syncthreads();
  for (int off = E_DIM / 2; off > 0; off >>= 1) {
    if (t < off) red[t] = red[t] + red[t + off];
    __syncthreads();
  }
  n_out[t] = (e / red[0]) * N_TOT;
}

// One wave32 per sample; lane owns cohorts j = i*32 + lane, i = 0..31.
__global__ __launch_bounds__(32) void sim_kernel(
    const float* __restrict__ xs, const float* __restrict__ n_ws,
    float* __restrict__ energies) {
  const int s = blockIdx.x;
  const int lane = threadIdx.x;

  float n[32], fn[32], x[32], x0[32];
#pragma unroll
  for (int i = 0; i < 32; ++i) {
    const int j = i * 32 + lane;
    const float nj = n_ws[j];
    n[i] = nj;
    fn[i] = FINAL_E * nj;
    const float v0 = xs[s * E_DIM + j] * nj;
    x0[i] = v0;
    x[i] = v0;
  }

  for (int k = 0; k < T_DIM; ++k) {
    float partial = 0.0f;
    bool act[32];
#pragma unroll
    for (int i = 0; i < 32; ++i) {
      const int j = i * 32 + lane;
      act[i] = (j <= k) && (x[i] < fn[i]);
      partial += act[i] ? n[i] : 0.0f;
    }
    // denom[s] = max(sum_j V*n, 1e-9): exact f32 wave reduction on the
    // matrix pipe (EXEC is all-ones here; conditionals above are selects).
    float denom = fmaxf(wave_sum_wmma(partial), 1e-9f);
    const float share = P_CAP / denom;
#pragma unroll
    for (int i = 0; i < 32; ++i) {
      // DECAY = 0  =>  rate limit is U0 * n
      float u = fminf(fminf(share * n[i], U0_RATE * n[i]), fn[i] - x[i]);
      x[i] += act[i] ? u : 0.0f;
    }
  }

  float esum = 0.0f;
#pragma unroll
  for (int i = 0; i < 32; ++i) esum += x[i] - x0[i];
  const float tot = wave_sum_wmma(esum);
  if (lane == 0) energies[s] = tot;   // after last WMMA: divergence is safe
}

// Fixed-order final mean (bit-deterministic across graph replays).
__global__ void finalize_kernel(const float* __restrict__ energies,
                                float* __restrict__ out, int S) {
  if (threadIdx.x == 0 && blockIdx.x == 0) {
    float acc = 0.0f;
    for (int s = 0; s < S; ++s) acc += energies[s];
    out[0] = -acc / (float)S;   // mean(-energy); balanced_factor = 0
  }
}

extern "C" void kernel_launch(void* const* d_in, const int* in_sizes, int n_in,
                              void* d_out, int out_size, void* d_ws,
                              size_t ws_size, hipStream_t stream) {
  const float* y = (const float*)d_in[0];    // (E,)   logits
  const float* xs = (const float*)d_in[1];   // (S, E) initial SoC draws
  const int S = in_sizes[1] / E_DIM;

  float* n_ws = (float*)d_ws;                // E floats
  float* energies = n_ws + E_DIM;            // S floats

  softmax_n_kernel<<<1, E_DIM, 0, stream>>>(y, n_ws);
  sim_kernel<<<S, 32, 0, stream>>>(xs, n_ws, energies);
  finalize_kernel<<<1, 32, 0, stream>>>(energies, (float*)d_out, S);
}